// PointNetPP_19576460936001
// MI455X (gfx1250) — compile-verified
//
#include <hip/hip_runtime.h>
#include <stdint.h>

// ---------------- problem constants (match reference) ----------------
#define BCLD  4
#define NPTS  4096
#define NS1   2048
#define NS2   512
#define KNBR  64
#define R1SQ  0.04f    // 0.2^2
#define R2SQ  0.16f    // 0.4^2

typedef float v2f __attribute__((ext_vector_type(2)));
typedef float v8f __attribute__((ext_vector_type(8)));
typedef unsigned int u32x4 __attribute__((ext_vector_type(4)));
typedef int i32x4 __attribute__((ext_vector_type(4)));
typedef int i32x8 __attribute__((ext_vector_type(8)));

// ---------------- CDNA5 TDM (tensor_load_to_lds) support ----------------
#if defined(__AMDGCN__) && __has_builtin(__builtin_amdgcn_tensor_load_to_lds)
#define HAVE_TDM 1
#else
#define HAVE_TDM 0
#endif

#if defined(__AMDGCN__) && __has_builtin(__builtin_amdgcn_s_wait_tensorcnt)
#define WAIT_TENSOR() __builtin_amdgcn_s_wait_tensorcnt(0)
#else
#define WAIT_TENSOR() ((void)0)
#endif

#if HAVE_TDM
// 1-D contiguous global->LDS DMA: nelem f32 elements (nelem <= 65535).
// D# layout per CDNA5 ISA ch.8 (group0 128b, group1 256b; groups 2/3 unused).
__device__ inline void tdm_load_1d_f32(const float* gsrc, uint32_t lds_byte_off,
                                       uint32_t nelem) {
  const uint64_t ga = (uint64_t)(uintptr_t)gsrc;
  u32x4 g0 = {0u, 0u, 0u, 0u};
  g0[0] = 1u;                                    // count=1 (valid user descriptor)
  g0[1] = lds_byte_off;                          // lds_addr (bytes)
  g0[2] = (uint32_t)ga;                          // global_addr[31:0]
  g0[3] = (uint32_t)((ga >> 32) & 0x01FFFFFFu)   // global_addr[56:32]
        | (2u << 30);                            // type = 2 ("image")
  i32x8 g1 = {0, 0, 0, 0, 0, 0, 0, 0};
  g1[0] = (int)(2u << 16);                       // data_size=2 (4 bytes); mask=0
  g1[1] = (int)((nelem & 0xffffu) << 16);        // tensor_dim0[15:0]
  g1[2] = (int)((nelem >> 16) & 0xffffu)         // tensor_dim0[31:16]
        | (int)(1u << 16);                       // tensor_dim1 = 1
  g1[3] = (int)((nelem & 0xffffu) << 16);        // tile_dim0 = nelem
  g1[4] = 0;                                     // tile_dim1=0, tile_dim2=0 (unused)
  g1[5] = (int)nelem;                            // tensor_dim0_stride low32
  g1[6] = 0;
  g1[7] = 0;
  i32x4 z4 = {0, 0, 0, 0};
#if __clang_major__ >= 23
  i32x8 z8 = {0, 0, 0, 0, 0, 0, 0, 0};
  __builtin_amdgcn_tensor_load_to_lds(g0, g1, z4, z4, z8, 0);
#else
  __builtin_amdgcn_tensor_load_to_lds(g0, g1, z4, z4, 0);
#endif
}
#endif

// ---------------- wave32 packed reductions ----------------
__device__ inline unsigned long long shfl_xor_u64(unsigned long long v, int m) {
  unsigned lo = __shfl_xor((unsigned)v, m, 32);
  unsigned hi = __shfl_xor((unsigned)(v >> 32), m, 32);
  return ((unsigned long long)hi << 32) | (unsigned long long)lo;
}
__device__ inline unsigned long long wave_max_u64(unsigned long long v) {
#pragma unroll
  for (int m = 16; m; m >>= 1) { unsigned long long o = shfl_xor_u64(v, m); v = (o > v) ? o : v; }
  return v;
}
__device__ inline unsigned long long wave_min_u64(unsigned long long v) {
#pragma unroll
  for (int m = 16; m; m >>= 1) { unsigned long long o = shfl_xor_u64(v, m); v = (o < v) ? o : v; }
  return v;
}

// ---------------- FPS: one 1024-thread workgroup per cloud ----------------
// Coordinates + running min-distances live in registers (<=4 pts/thread).
// jnp.argmax tie-break = first index -> pack (bits<<32)|~i, take max.
__global__ void fps_kernel(const float* __restrict__ pos, int n, int nsamp,
                           int* __restrict__ sidx, float* __restrict__ qpos) {
  __shared__ unsigned long long red[33];   // [0..31] per-wave, [32] winner
  __shared__ float bc[3];                  // winner coords broadcast

  const int b = blockIdx.x;
  const int tid = threadIdx.x;
  const int nt = blockDim.x;               // 1024
  const int lane = tid & 31, wave = tid >> 5, nwaves = nt >> 5;
  const int slots = n / nt;                // 4 (pass 1) or 2 (pass 2)

  float pxr[4], pyr[4], pzr[4], ddr[4];
  const float* P = pos + (size_t)b * n * 3;
  for (int s = 0; s < slots; ++s) {
    const int i = tid + s * nt;
    pxr[s] = P[i * 3 + 0]; pyr[s] = P[i * 3 + 1]; pzr[s] = P[i * 3 + 2];
  }
  if (tid == 0) {
    bc[0] = pxr[0]; bc[1] = pyr[0]; bc[2] = pzr[0];
    sidx[(size_t)b * nsamp] = 0;
    float* qp = qpos + (size_t)b * nsamp * 3;
    qp[0] = pxr[0]; qp[1] = pyr[0]; qp[2] = pzr[0];
  }
  __syncthreads();
  {
    const float x0 = bc[0], y0 = bc[1], z0 = bc[2];
    for (int s = 0; s < slots; ++s) {
      float dx = pxr[s] - x0, dy = pyr[s] - y0, dz = pzr[s] - z0;
      ddr[s] = dx * dx + dy * dy + dz * dz;
    }
  }
  __syncthreads();

  for (int st = 1; st < nsamp; ++st) {
    unsigned long long best = 0ULL;
    for (int s = 0; s < slots; ++s) {
      const unsigned i = (unsigned)(tid + s * nt);
      unsigned long long p =
          ((unsigned long long)__float_as_uint(ddr[s]) << 32) | (unsigned)(~i);
      if (p > best) best = p;
    }
    best = wave_max_u64(best);
    if (lane == 0) red[wave] = best;
    __syncthreads();                                       // A
    if (tid == 0) {
      unsigned long long v = red[0];
      for (int w = 1; w < nwaves; ++w) if (red[w] > v) v = red[w];
      red[32] = v;
    }
    __syncthreads();                                       // B
    const int j = (int)(~(unsigned)(red[32] & 0xffffffffu));
    if (tid == (j % nt)) {                                 // owner broadcasts
      const int sl = j / nt;
      bc[0] = pxr[sl]; bc[1] = pyr[sl]; bc[2] = pzr[sl];
      sidx[(size_t)b * nsamp + st] = j;
      float* qp = qpos + ((size_t)b * nsamp + st) * 3;
      qp[0] = pxr[sl]; qp[1] = pyr[sl]; qp[2] = pzr[sl];
    }
    __syncthreads();                                       // C
    const float jx = bc[0], jy = bc[1], jz = bc[2];
    for (int s = 0; s < slots; ++s) {
      float dx = pxr[s] - jx, dy = pyr[s] - jy, dz = pzr[s] - jz;
      float nd = dx * dx + dy * dy + dz * dz;
      if (nd < ddr[s]) ddr[s] = nd;
    }
  }
}

// ---------------- radius-kNN: one 256-thread block per query ----------------
// Per-thread candidates in registers; each round block-reduces 256 minima,
// winner's owner rescans its <=16 slots. Ties -> lowest index (= lax.top_k).
__global__ void knn_kernel(const float* __restrict__ src, const float* __restrict__ q,
                           int ns, int m, float r2,
                           int* __restrict__ nbr, int* __restrict__ cnt) {
  __shared__ unsigned long long red[9];    // [0..7] per-wave, [8] winner

  const int b = blockIdx.x / m;
  const int mi = blockIdx.x % m;
  const int tid = threadIdx.x, ntr = blockDim.x;           // 256
  const int lane = tid & 31, wave = tid >> 5, nwaves = ntr >> 5;
  const int slots = ns / ntr;                              // 16 or 8

  const float* Q = q + ((size_t)b * m + mi) * 3;
  const float qx = Q[0], qy = Q[1], qz = Q[2];
  const float* S = src + (size_t)b * ns * 3;

  float myd[16];
  unsigned long long mybest = ~0ULL;
  for (int s = 0; s < slots; ++s) {
    const int i = tid + s * ntr;
    float dx = S[i * 3 + 0] - qx, dy = S[i * 3 + 1] - qy, dz = S[i * 3 + 2] - qz;
    myd[s] = dx * dx + dy * dy + dz * dz;
    unsigned long long p =
        ((unsigned long long)__float_as_uint(myd[s]) << 32) | (unsigned)i;
    if (p < mybest) mybest = p;
  }

  int* NB = nbr + ((size_t)b * m + mi) * KNBR;
  int found = 0;
  for (int k = 0; k < KNBR; ++k) {
    unsigned long long best = wave_min_u64(mybest);
    if (lane == 0) red[wave] = best;
    __syncthreads();
    if (tid == 0) {
      unsigned long long v = red[0];
      for (int w = 1; w < nwaves; ++w) if (red[w] < v) v = red[w];
      red[8] = v;
    }
    __syncthreads();
    const unsigned long long v = red[8];
    const float dist = __uint_as_float((unsigned)(v >> 32));
    const int j = (int)(unsigned)(v & 0xffffffffu);
    if (dist <= r2) {
      if (tid == 0) NB[k] = j;
      if (tid == (j % ntr)) {               // owner removes winner, rescans
        myd[j / ntr] = __uint_as_float(0x7f800000u);
        mybest = ~0ULL;
        for (int s = 0; s < slots; ++s) {
          unsigned long long p =
              ((unsigned long long)__float_as_uint(myd[s]) << 32) |
              (unsigned)(tid + s * ntr);
          if (p < mybest) mybest = p;
        }
      }
      ++found;
    } else {
      if (tid == 0) { for (int kk = k; kk < KNBR; ++kk) NB[kk] = 0; }
      break;
    }
  }
  if (tid == 0) cnt[(size_t)b * m + mi] = found;
}

// ---------------- WMMA f32 16x16x4 MLP layer ----------------
// in : LDS activations [rows][kinp] (pad cols zeroed), rows multiple of 16
// W  : row-major [kin][nout] (global or LDS), bias[nout]; out : LDS [rows][nout]
__device__ inline void mlp_layer(const float* __restrict__ in, int rows, int kin, int kinp,
                                 const float* __restrict__ W, const float* __restrict__ bias,
                                 int nout, float* __restrict__ out, bool relu) {
  const int lane = threadIdx.x & 31;
  const int wave = threadIdx.x >> 5;
  const int nwaves = blockDim.x >> 5;
  const int half = lane >> 4;          // A/B: lanes16-31 carry K+2..K+3
  const int l = lane & 15;
  const int nM = rows >> 4, nN = nout >> 4;
  for (int t = wave; t < nM * nN; t += nwaves) {
    const int mt = t / nN, ntl = t % nN;
    const float* arow = in + (size_t)(mt * 16 + l) * kinp;
    const int col = ntl * 16 + l;
    v8f acc = {0.f, 0.f, 0.f, 0.f, 0.f, 0.f, 0.f, 0.f};
    for (int k = 0; k < kinp; k += 4) {
      const int c0 = k + (half ? 2 : 0);
      v2f a, bfrag;
      a.x = arow[c0];
      a.y = arow[c0 + 1];
      bfrag.x = (c0 < kin)     ? W[(size_t)c0 * nout + col]       : 0.f;
      bfrag.y = (c0 + 1 < kin) ? W[(size_t)(c0 + 1) * nout + col] : 0.f;
      acc = __builtin_amdgcn_wmma_f32_16x16x4_f32(false, a, false, bfrag,
                                                  (short)0, acc, false, false);
    }
    const float bv = bias[col];
#pragma unroll
    for (int j = 0; j < 8; ++j) {
      float v = acc[j] + bv;
      if (relu) v = fmaxf(v, 0.f);
      out[(size_t)(mt * 16 + half * 8 + j) * nout + col] = v;
    }
  }
  __syncthreads();
}

// Final layer variant: no LDS store; per-column max over all rows -> atomicMax(global).
__device__ inline void mlp_layer_gmax(const float* __restrict__ in, int rows, int kin, int kinp,
                                      const float* __restrict__ W, const float* __restrict__ bias,
                                      int nout, float* __restrict__ gout) {
  const int lane = threadIdx.x & 31;
  const int wave = threadIdx.x >> 5;
  const int nwaves = blockDim.x >> 5;
  const int half = lane >> 4;
  const int l = lane & 15;
  const int nM = rows >> 4, nN = nout >> 4;
  for (int t = wave; t < nM * nN; t += nwaves) {
    const int ntl = t % nN;
    const int mt = t / nN;
    const float* arow = in + (size_t)(mt * 16 + l) * kinp;
    const int col = ntl * 16 + l;
    v8f acc = {0.f, 0.f, 0.f, 0.f, 0.f, 0.f, 0.f, 0.f};
    for (int k = 0; k < kinp; k += 4) {
      const int c0 = k + (half ? 2 : 0);
      v2f a, bfrag;
      a.x = arow[c0];
      a.y = arow[c0 + 1];
      bfrag.x = (c0 < kin)     ? W[(size_t)c0 * nout + col]       : 0.f;
      bfrag.y = (c0 + 1 < kin) ? W[(size_t)(c0 + 1) * nout + col] : 0.f;
      acc = __builtin_amdgcn_wmma_f32_16x16x4_f32(false, a, false, bfrag,
                                                  (short)0, acc, false, false);
    }
    const float bv = bias[col];
    float mx = acc[0] + bv;
#pragma unroll
    for (int j = 1; j < 8; ++j) mx = fmaxf(mx, acc[j] + bv);
    mx = fmaxf(mx, __shfl_xor(mx, 16, 32));   // combine the two 8-row halves
    if (half == 0) atomicMax(&gout[col], mx);
  }
  __syncthreads();
}

// ---------------- SA1 fused MLP: one block (4 waves) per query ----------------
// W2/W3 staged into LDS via TDM, overlapped with gather + layer-1 WMMAs.
__global__ void sa1_kernel(const float* __restrict__ pos, const float* __restrict__ qpos,
                           const int* __restrict__ nbr, const int* __restrict__ cnt,
                           const float* __restrict__ W1, const float* __restrict__ b1,
                           const float* __restrict__ W2, const float* __restrict__ b2,
                           const float* __restrict__ W3, const float* __restrict__ b3,
                           float* __restrict__ x1, int n, int m) {
  extern __shared__ float sm[];
  float* bufA = sm;                       // 64*64 floats (also input [64][4])
  float* bufB = sm + 64 * 64;             // 64*128 floats
  float* w2s  = sm + 64 * 64 + 64 * 128;  // 64*64 floats
  float* w3s  = w2s + 64 * 64;            // 64*128 floats
  const int b = blockIdx.x / m, mi = blockIdx.x % m;
  const int* NB = nbr + ((size_t)b * m + mi) * KNBR;
  const float* Q = qpos + ((size_t)b * m + mi) * 3;
  const float qx = Q[0], qy = Q[1], qz = Q[2];
  const float* P = pos + (size_t)b * n * 3;

#if HAVE_TDM
  if (threadIdx.x < 32) {                 // wave 0 issues both DMAs
    tdm_load_1d_f32(W2, (uint32_t)((const char*)w2s - (const char*)sm), 64 * 64);
    tdm_load_1d_f32(W3, (uint32_t)((const char*)w3s - (const char*)sm), 64 * 128);
  }
#else
  for (int e = threadIdx.x; e < 64 * 64; e += blockDim.x) w2s[e] = W2[e];
  for (int e = threadIdx.x; e < 64 * 128; e += blockDim.x) w3s[e] = W3[e];
#endif

  for (int r = threadIdx.x; r < KNBR; r += blockDim.x) {
    const int j = NB[r];
    bufA[r * 4 + 0] = P[j * 3 + 0] - qx;
    bufA[r * 4 + 1] = P[j * 3 + 1] - qy;
    bufA[r * 4 + 2] = P[j * 3 + 2] - qz;
    bufA[r * 4 + 3] = 0.f;
  }
  __syncthreads();
  mlp_layer(bufA, 64, 3, 4, W1, b1, 64, bufB, true);   // W1 tiny: read from L2
  WAIT_TENSOR();                                        // issuing wave drains TDM
  __syncthreads();                                      // others see LDS weights
  mlp_layer(bufB, 64, 64, 64, w2s, b2, 64, bufA, true);
  mlp_layer(bufA, 64, 64, 64, w3s, b3, 128, bufB, false);
  const int c = cnt[(size_t)b * m + mi];
  for (int col = threadIdx.x; col < 128; col += blockDim.x) {
    float mx = -INFINITY;
    for (int r = 0; r < c; ++r) mx = fmaxf(mx, bufB[r * 128 + col]);
    x1[((size_t)b * m + mi) * 128 + col] = (c > 0) ? mx : 0.f;
  }
}

// ---------------- SA2 fused MLP: one block per query ----------------
__global__ void sa2_kernel(const float* __restrict__ x1, const float* __restrict__ srcpos,
                           const float* __restrict__ qpos,
                           const int* __restrict__ nbr, const int* __restrict__ cnt,
                           const float* __restrict__ W1, const float* __restrict__ b1,
                           const float* __restrict__ W2, const float* __restrict__ b2,
                           const float* __restrict__ W3, const float* __restrict__ b3,
                           float* __restrict__ x2, int ns, int m) {
  extern __shared__ float sm[];
  float* bufA = sm;              // 64*132 floats
  float* bufB = sm + 64 * 132;   // 64*256 floats
  const int b = blockIdx.x / m, mi = blockIdx.x % m;
  const int* NB = nbr + ((size_t)b * m + mi) * KNBR;
  const float* Q = qpos + ((size_t)b * m + mi) * 3;
  for (int e = threadIdx.x; e < 64 * 132; e += blockDim.x) {
    const int r = e / 132, c = e % 132;
    const int j = NB[r];
    float v;
    if (c < 128)      v = x1[((size_t)b * ns + j) * 128 + c];
    else if (c < 131) v = srcpos[((size_t)b * ns + j) * 3 + (c - 128)] - Q[c - 128];
    else              v = 0.f;
    bufA[e] = v;
  }
  __syncthreads();
  mlp_layer(bufA, 64, 131, 132, W1, b1, 128, bufB, true);
  mlp_layer(bufB, 64, 128, 128, W2, b2, 128, bufA, true);
  mlp_layer(bufA, 64, 128, 128, W3, b3, 256, bufB, false);
  const int c = cnt[(size_t)b * m + mi];
  for (int col = threadIdx.x; col < 256; col += blockDim.x) {
    float mx = -INFINITY;
    for (int r = 0; r < c; ++r) mx = fmaxf(mx, bufB[r * 256 + col]);
    x2[((size_t)b * m + mi) * 256 + col] = (c > 0) ? mx : 0.f;
  }
}

// ---------------- g = -inf init ----------------
__global__ void ginit_kernel(float* __restrict__ g, int n) {
  const int i = blockIdx.x * blockDim.x + threadIdx.x;
  if (i < n) g[i] = -INFINITY;
}

// ---------------- MLP3 + global max pool: 64 rows per block ----------------
__global__ void mlp3_kernel(const float* __restrict__ x2, const float* __restrict__ p2,
                            const float* __restrict__ W1, const float* __restrict__ b1,
                            const float* __restrict__ W2, const float* __restrict__ b2,
                            const float* __restrict__ W3, const float* __restrict__ b3,
                            float* __restrict__ g) {
  extern __shared__ float sm[];
  float* bufA = sm;              // 64*512 floats (input uses stride 260)
  float* bufB = sm + 64 * 512;   // 64*256 floats
  __builtin_prefetch(W1, 0, 0);
  __builtin_prefetch(W2, 0, 0);
  __builtin_prefetch(W3, 0, 0);
  const int b = blockIdx.x / 8;
  const int chunk = blockIdx.x % 8;
  for (int e = threadIdx.x; e < 64 * 260; e += blockDim.x) {
    const int r = e / 260, c = e % 260;
    const size_t gr = (size_t)b * NS2 + chunk * 64 + r;
    float v;
    if (c < 256)      v = x2[gr * 256 + c];
    else if (c < 259) v = p2[gr * 3 + (c - 256)];
    else              v = 0.f;
    bufA[r * 260 + c] = v;
  }
  __syncthreads();
  mlp_layer(bufA, 64, 259, 260, W1, b1, 256, bufB, true);
  mlp_layer(bufB, 64, 256, 256, W2, b2, 512, bufA, true);
  mlp_layer_gmax(bufA, 64, 512, 512, W3, b3, 1024, g + (size_t)b * 1024);
}

// ---------------- head MLP: rows padded 4 -> 16 ----------------
__global__ void head_kernel(const float* __restrict__ g,
                            const float* __restrict__ W1, const float* __restrict__ b1,
                            const float* __restrict__ W2, const float* __restrict__ b2,
                            float* __restrict__ out) {
  extern __shared__ float sm[];
  float* bufA = sm;               // 16*1024 floats
  float* bufB = sm + 16 * 1024;   // 16*512 floats
  __builtin_prefetch(W1, 0, 0);
  __builtin_prefetch(W2, 0, 0);
  for (int e = threadIdx.x; e < 16 * 1024; e += blockDim.x) {
    const int r = e / 1024, c = e % 1024;
    bufA[e] = (r < BCLD) ? g[(size_t)r * 1024 + c] : 0.f;
  }
  __syncthreads();
  mlp_layer(bufA, 16, 1024, 1024, W1, b1, 512, bufB, true);
  mlp_layer(bufB, 16, 512, 512, W2, b2, 256, bufA, false);
  for (int e = threadIdx.x; e < BCLD * 256; e += blockDim.x) out[e] = bufA[e];
}

// ---------------- launch ----------------
extern "C" void kernel_launch(void* const* d_in, const int* in_sizes, int n_in,
                              void* d_out, int out_size, void* d_ws, size_t ws_size,
                              hipStream_t stream) {
  (void)in_sizes; (void)n_in; (void)out_size; (void)ws_size;
  const float* pos  = (const float*)d_in[0];
  const float* m1W1 = (const float*)d_in[1];  const float* m1b1 = (const float*)d_in[2];
  const float* m1W2 = (const float*)d_in[3];  const float* m1b2 = (const float*)d_in[4];
  const float* m1W3 = (const float*)d_in[5];  const float* m1b3 = (const float*)d_in[6];
  const float* m2W1 = (const float*)d_in[7];  const float* m2b1 = (const float*)d_in[8];
  const float* m2W2 = (const float*)d_in[9];  const float* m2b2 = (const float*)d_in[10];
  const float* m2W3 = (const float*)d_in[11]; const float* m2b3 = (const float*)d_in[12];
  const float* m3W1 = (const float*)d_in[13]; const float* m3b1 = (const float*)d_in[14];
  const float* m3W2 = (const float*)d_in[15]; const float* m3b2 = (const float*)d_in[16];
  const float* m3W3 = (const float*)d_in[17]; const float* m3b3 = (const float*)d_in[18];
  const float* hW1  = (const float*)d_in[19]; const float* hb1  = (const float*)d_in[20];
  const float* hW2  = (const float*)d_in[21]; const float* hb2  = (const float*)d_in[22];
  float* out = (float*)d_out;

  char* ws = (char*)d_ws;
  size_t off = 0;
  auto carve = [&](size_t bytes) -> void* {
    void* p = ws + off;
    off = (off + bytes + 255) & ~(size_t)255;
    return p;
  };
  int*   sidx1 = (int*)  carve((size_t)BCLD * NS1 * 4);
  float* q1    = (float*)carve((size_t)BCLD * NS1 * 3 * 4);
  int*   nbr1  = (int*)  carve((size_t)BCLD * NS1 * KNBR * 4);
  int*   cnt1  = (int*)  carve((size_t)BCLD * NS1 * 4);
  float* x1    = (float*)carve((size_t)BCLD * NS1 * 128 * 4);
  int*   sidx2 = (int*)  carve((size_t)BCLD * NS2 * 4);
  float* q2    = (float*)carve((size_t)BCLD * NS2 * 3 * 4);
  int*   nbr2  = (int*)  carve((size_t)BCLD * NS2 * KNBR * 4);
  int*   cnt2  = (int*)  carve((size_t)BCLD * NS2 * 4);
  float* x2    = (float*)carve((size_t)BCLD * NS2 * 256 * 4);
  float* g     = (float*)carve((size_t)BCLD * 1024 * 4);

  // Large dynamic-LDS opt-ins (host-side API, not stream ops; capture-safe)
  (void)hipFuncSetAttribute((const void*)sa1_kernel,  hipFuncAttributeMaxDynamicSharedMemorySize, 128 * 1024);
  (void)hipFuncSetAttribute((const void*)sa2_kernel,  hipFuncAttributeMaxDynamicSharedMemorySize, 128 * 1024);
  (void)hipFuncSetAttribute((const void*)mlp3_kernel, hipFuncAttributeMaxDynamicSharedMemorySize, 224 * 1024);
  (void)hipFuncSetAttribute((const void*)head_kernel, hipFuncAttributeMaxDynamicSharedMemorySize, 128 * 1024);

  const size_t sa1_lds  = (size_t)(64 * 64 + 64 * 128 + 64 * 64 + 64 * 128) * 4; // 96 KB
  const size_t sa2_lds  = (size_t)(64 * 132 + 64 * 256) * 4;                     // 97 KB
  const size_t mlp3_lds = (size_t)(64 * 512 + 64 * 256) * 4;                     // 192 KB
  const size_t head_lds = (size_t)(16 * 1024 + 16 * 512) * 4;                    // 96 KB

  // SA module 1
  fps_kernel<<<BCLD, 1024, 0, stream>>>(pos, NPTS, NS1, sidx1, q1);
  knn_kernel<<<BCLD * NS1, 256, 0, stream>>>(pos, q1, NPTS, NS1, R1SQ, nbr1, cnt1);
  sa1_kernel<<<BCLD * NS1, 128, sa1_lds, stream>>>(pos, q1, nbr1, cnt1,
                                                   m1W1, m1b1, m1W2, m1b2, m1W3, m1b3,
                                                   x1, NPTS, NS1);
  // SA module 2
  fps_kernel<<<BCLD, 1024, 0, stream>>>(q1, NS1, NS2, sidx2, q2);
  knn_kernel<<<BCLD * NS2, 256, 0, stream>>>(q1, q2, NS1, NS2, R2SQ, nbr2, cnt2);
  sa2_kernel<<<BCLD * NS2, 128, sa2_lds, stream>>>(x1, q1, q2, nbr2, cnt2,
                                                   m2W1, m2b1, m2W2, m2b2, m2W3, m2b3,
                                                   x2, NS1, NS2);
  // Global MLP + max pool
  ginit_kernel<<<(BCLD * 1024 + 255) / 256, 256, 0, stream>>>(g, BCLD * 1024);
  mlp3_kernel<<<BCLD * 8, 128, mlp3_lds, stream>>>(x2, q2,
                                                   m3W1, m3b1, m3W2, m3b2, m3W3, m3b3, g);
  // Head
  head_kernel<<<1, 128, head_lds, stream>>>(g, hW1, hb1, hW2, hb2, out);
}